// EncodingLayer_47261820125416
// MI455X (gfx1250) — compile-verified
//
#include <hip/hip_runtime.h>
#include <math.h>

typedef __attribute__((ext_vector_type(2))) float v2f;
typedef __attribute__((ext_vector_type(4))) float v4f;
typedef __attribute__((ext_vector_type(8))) float v8f;

#define B_    4
#define S_    512
#define D_    64
#define H_    8
#define DH_   8
#define DFF_  256
#define NTOK  (B_ * S_)      /* 2048 */
#define NELEM (NTOK * D_)    /* 131072 */

__device__ __forceinline__ v8f wmma4(v2f a, v2f b, v8f c) {
  // V_WMMA_F32_16X16X4_F32 : D(16x16 f32) = A(16x4 f32) * B(4x16 f32) + C
  return __builtin_amdgcn_wmma_f32_16x16x4_f32(false, a, false, b, (short)0, c,
                                               false, false);
}

// CDNA5 async direct-to-LDS copy (GLOBAL_LOAD_ASYNC_TO_LDS_B128, ASYNCcnt).
// Bypasses VGPRs: 16 bytes per lane from global to LDS.
__device__ __forceinline__ void async_lds_b128(unsigned ldsOff,
                                               const float* g) {
  unsigned long long ga = (unsigned long long)(size_t)g;
  asm volatile("global_load_async_to_lds_b128 %0, %1, off"
               :
               : "v"(ldsOff), "v"(ga)
               : "memory");
}
__device__ __forceinline__ void wait_async0() {
  asm volatile("s_wait_asynccnt 0" ::: "memory");
}

// ---------------------------------------------------------------------------
// Shared GEMM body:  C[M x N] = A[M x K] @ W[K x N]  (+ epilogue)
//   mode 0: C = AW + bias
//   mode 1: C = AW + bias ; Z = C + Y        (attn proj + residual input z)
//   mode 2: C = relu(AW + bias)              (FFN layer 1)
//   mode 3: C = AW + bias + Y                (FFN layer 2 + att residual)
// block = 256 (8 waves); each wave owns a 16x64 band of a 128x64 tile.
// W tile staged to LDS via async DMA; A pairs fully unrolled in VGPRs.
// ---------------------------------------------------------------------------
__device__ __forceinline__ void gemm_body(
    const float* __restrict__ A, const float* __restrict__ W,
    const float* __restrict__ bias, float* __restrict__ C,
    int K, int N, int mode, const float* __restrict__ Y,
    float* __restrict__ Z, float* Wl, int rowBlk, int nbase) {
  const int tid  = threadIdx.x;
  const int wave = tid >> 5, lane = tid & 31;
  const int lo = lane & 15, hi = lane >> 4;
  const int row0 = rowBlk * 128 + wave * 16;

  v8f acc[4] = {};  // 4 column tiles of 16

  for (int kc = 0; kc < K; kc += 64) {
    // --- async-DMA the 64x64 W tile into LDS (1024 b128 chunks, 4/thread) ---
#pragma unroll
    for (int i = 0; i < 4; ++i) {
      const int j  = tid + i * 256;        // 0..1023
      const int kk = j >> 4;               // row within tile
      const int cb = (j & 15) * 4;         // 4-float column chunk
      async_lds_b128((unsigned)(size_t)&Wl[kk * 64 + cb],
                     &W[(size_t)(kc + kk) * N + nbase + cb]);
    }
    // --- hoist this wave's 16 A pairs so global loads pipeline ---
    const float* Arow = A + (size_t)(row0 + lo) * K + kc;
    v2f areg[16];
#pragma unroll
    for (int kb = 0; kb < 16; ++kb) {
      const int kk = kb * 4 + 2 * hi;
      areg[kb].x = Arow[kk];
      areg[kb].y = Arow[kk + 1];
    }
    wait_async0();
    __syncthreads();
    // FULL unroll: constant areg indices keep the array in direct VGPRs
#pragma unroll
    for (int kb = 0; kb < 16; ++kb) {
      const int kk = kb * 4 + 2 * hi;  // k index pair this half-wave owns
#pragma unroll
      for (int nt = 0; nt < 4; ++nt) {
        v2f b;
        b.x = Wl[kk * 64 + nt * 16 + lo];
        b.y = Wl[(kk + 1) * 64 + nt * 16 + lo];
        acc[nt] = wmma4(areg[kb], b, acc[nt]);
      }
    }
    __syncthreads();
  }

#pragma unroll
  for (int nt = 0; nt < 4; ++nt) {
    const int col = nbase + nt * 16 + lo;
    const float bval = bias[col];
#pragma unroll
    for (int v = 0; v < 8; ++v) {
      const int row = row0 + v + 8 * hi;  // C layout: m = v + 8*hi, n = lo
      const size_t idx = (size_t)row * N + col;
      float val = acc[nt][v] + bval;
      if (mode == 2) val = fmaxf(val, 0.0f);
      if (mode == 3) val += Y[idx];
      C[idx] = val;
      if (mode == 1) Z[idx] = val + Y[idx];
    }
  }
}

__global__ __launch_bounds__(256) void gemm_wmma_k(
    const float* __restrict__ A, const float* __restrict__ W,
    const float* __restrict__ bias, float* __restrict__ C,
    int K, int N, int mode, const float* __restrict__ Y,
    float* __restrict__ Z) {
  __shared__ float Wl[64 * 64];
  gemm_body(A, W, bias, C, K, N, mode, Y, Z, Wl, blockIdx.x,
            blockIdx.y * 64);
}

// Q/K/V fused: one launch, gridDim.z = 3 selects projection (saves launches
// on the serial RK dependency chain).
__global__ __launch_bounds__(256) void gemm_qkv_k(
    const float* __restrict__ A, const float* __restrict__ Wq,
    const float* __restrict__ Wk, const float* __restrict__ Wv,
    const float* __restrict__ bq, const float* __restrict__ bk,
    const float* __restrict__ bv, float* __restrict__ Cq,
    float* __restrict__ Ck, float* __restrict__ Cv) {
  __shared__ float Wl[64 * 64];
  const int z = blockIdx.z;
  const float* W = (z == 0) ? Wq : (z == 1) ? Wk : Wv;
  const float* b = (z == 0) ? bq : (z == 1) ? bk : bv;
  float* C = (z == 0) ? Cq : (z == 1) ? Ck : Cv;
  gemm_body(A, W, b, C, 64, 64, 0, nullptr, nullptr, Wl, blockIdx.x, 0);
}

// ---------------------------------------------------------------------------
// Flash attention: one block per (batch, head); 8 waves, each wave streams
// 4 query tiles of 16 rows over all 32 key tiles with online softmax.
// Q/K/V stay in [B,S,64] layout (head h = columns h*8 .. h*8+7).
// Note: reference mask [B,1,S,1] broadcasts over the KEY axis (uniform row
// shift) — mathematically cancels in softmax, but applied anyway.
// ---------------------------------------------------------------------------
__global__ __launch_bounds__(256) void attn_wmma_k(
    const float* __restrict__ Q, const float* __restrict__ Kb,
    const float* __restrict__ Vb, const float* __restrict__ mask,
    float* __restrict__ O) {
  __shared__ float pbuf[8 * 256];  // per-wave 16x16 P transpose buffer
  const int tid  = threadIdx.x;
  const int wave = tid >> 5, lane = tid & 31;
  const int lo = lane & 15, hi = lane >> 4;
  const int b = blockIdx.x >> 3, h = blockIdx.x & 7;
  const float scale = 0.35355339059327378f;  // 1/sqrt(8)
  float* pw = pbuf + wave * 256;
  const size_t base = (size_t)b * S_ * D_;

  for (int qt = wave; qt < S_ / 16; qt += 8) {
    const int q0 = qt * 16;
    // Q tile in A layout (dh=8 -> two 16x4 k-blocks)
    v2f aq[2];
#pragma unroll
    for (int kb = 0; kb < 2; ++kb) {
      const int kd = h * DH_ + kb * 4 + 2 * hi;
      aq[kb].x = Q[base + (size_t)(q0 + lo) * D_ + kd];
      aq[kb].y = Q[base + (size_t)(q0 + lo) * D_ + kd + 1];
    }
    float rmask[8], mrun[8], lrun[8];
#pragma unroll
    for (int v = 0; v < 8; ++v) {
      rmask[v] = mask[b * S_ + q0 + v + 8 * hi] * (-1e9f);
      mrun[v] = -1e30f;
      lrun[v] = 0.0f;
    }
    v8f accO = {};

    for (int kt = 0; kt < S_ / 16; ++kt) {
      const int k0 = kt * 16;
      // scores S(16q x 16k) = Qtile @ Ktile^T  (B = K^T, element (kd, key))
      v8f s = {};
#pragma unroll
      for (int kb = 0; kb < 2; ++kb) {
        const int kd = h * DH_ + kb * 4 + 2 * hi;
        v2f bk;
        bk.x = Kb[base + (size_t)(k0 + lo) * D_ + kd];
        bk.y = Kb[base + (size_t)(k0 + lo) * D_ + kd + 1];
        s = wmma4(aq[kb], bk, s);
      }
      // online softmax: rows striped m = v + 8*hi, row spans 16 lanes
      float p[8];
#pragma unroll
      for (int v = 0; v < 8; ++v) {
        float sv = s[v] * scale + rmask[v];
        float mx = sv;
        mx = fmaxf(mx, __shfl_xor(mx, 1, 16));
        mx = fmaxf(mx, __shfl_xor(mx, 2, 16));
        mx = fmaxf(mx, __shfl_xor(mx, 4, 16));
        mx = fmaxf(mx, __shfl_xor(mx, 8, 16));
        const float mnew  = fmaxf(mrun[v], mx);
        const float alpha = __expf(mrun[v] - mnew);
        p[v] = __expf(sv - mnew);
        float rs = p[v];
        rs += __shfl_xor(rs, 1, 16);
        rs += __shfl_xor(rs, 2, 16);
        rs += __shfl_xor(rs, 4, 16);
        rs += __shfl_xor(rs, 8, 16);
        lrun[v] = lrun[v] * alpha + rs;
        mrun[v] = mnew;
        accO[v] = accO[v] * alpha;
        pw[(v + 8 * hi) * 16 + lo] = p[v];  // C layout -> row-major 16x16
      }
      // O(16q x 8d) += P(16x16) @ Vtile(16k x 8d); K=16 -> 4 wmma steps
#pragma unroll
      for (int kb = 0; kb < 4; ++kb) {
        const int kk = kb * 4 + 2 * hi;
        v2f pa, bv;
        pa.x = pw[lo * 16 + kk];      // A layout: m = lo, k = kk(+1)
        pa.y = pw[lo * 16 + kk + 1];
        const int vc = h * DH_ + (lo & 7);  // cols 8..15 duplicated (unused)
        bv.x = Vb[base + (size_t)(k0 + kk) * D_ + vc];
        bv.y = Vb[base + (size_t)(k0 + kk + 1) * D_ + vc];
        accO = wmma4(pa, bv, accO);
      }
    }
    if (lo < DH_) {
#pragma unroll
      for (int v = 0; v < 8; ++v)
        O[base + (size_t)(q0 + v + 8 * hi) * D_ + h * DH_ + lo] =
            accO[v] / lrun[v];
    }
  }
}

// ---------------------------------------------------------------------------
// RK linear combination (float4-vectorized): out = base + sum(ci * pi)
// (zero coefficients skip their load -> safe on poisoned scratch)
// ---------------------------------------------------------------------------
__global__ __launch_bounds__(256) void combo_k(
    v4f* __restrict__ out, const v4f* __restrict__ base,
    const v4f* __restrict__ p0, const v4f* __restrict__ p1,
    const v4f* __restrict__ p2, const v4f* __restrict__ p3,
    const v4f* __restrict__ p4, const v4f* __restrict__ p5,
    float c0, float c1, float c2, float c3, float c4, float c5, int n4) {
  const int i = blockIdx.x * 256 + threadIdx.x;
  if (i >= n4) return;
  v4f acc = base[i];
  if (c0 != 0.0f) acc += p0[i] * c0;
  if (c1 != 0.0f) acc += p1[i] * c1;
  if (c2 != 0.0f) acc += p2[i] * c2;
  if (c3 != 0.0f) acc += p3[i] * c3;
  if (c4 != 0.0f) acc += p4[i] * c4;
  if (c5 != 0.0f) acc += p5[i] * c5;
  out[i] = acc;
}

// ---------------------------------------------------------------------------
extern "C" void kernel_launch(void* const* d_in, const int* in_sizes, int n_in,
                              void* d_out, int out_size, void* d_ws,
                              size_t ws_size, hipStream_t stream) {
  (void)in_sizes; (void)n_in; (void)out_size; (void)ws_size;
  const float* x    = (const float*)d_in[0];
  const float* mask = (const float*)d_in[1];
  const float* Wq = (const float*)d_in[2];  const float* bq = (const float*)d_in[3];
  const float* Wk = (const float*)d_in[4];  const float* bk = (const float*)d_in[5];
  const float* Wv = (const float*)d_in[6];  const float* bv = (const float*)d_in[7];
  const float* Wo = (const float*)d_in[8];  const float* bo = (const float*)d_in[9];
  const float* W1 = (const float*)d_in[10]; const float* b1 = (const float*)d_in[11];
  const float* W2 = (const float*)d_in[12]; const float* b2 = (const float*)d_in[13];

  float* ws = (float*)d_ws;
  const int N = NELEM;
  float* y = ws;
  float* kst[6];
  for (int i = 0; i < 6; ++i) kst[i] = ws + (size_t)N * (1 + i);
  float* ystage = ws + (size_t)N * 7;
  float* qb     = ws + (size_t)N * 8;
  float* kbuf   = ws + (size_t)N * 9;
  float* vbuf   = ws + (size_t)N * 10;
  float* attnO  = ws + (size_t)N * 11;
  float* att    = ws + (size_t)N * 12;
  float* zb     = ws + (size_t)N * 13;
  float* h1     = ws + (size_t)N * 14;  // B*S*256 = 4N floats

  const dim3 blk(256);
  const dim3 gP(NTOK / 128, 1);      // 64-wide GEMMs
  const dim3 gQKV(NTOK / 128, 1, 3); // fused Q/K/V
  const dim3 gF1(NTOK / 128, 4);     // N=256 GEMM
  const dim3 gA(B_ * H_);
  const int n4 = N / 4;
  const int cgrid = (n4 + 255) / 256;

  auto combo = [&](float* o, const float* bse, float* const* p,
                   const float* c) {
    combo_k<<<cgrid, blk, 0, stream>>>(
        (v4f*)o, (const v4f*)bse, (const v4f*)p[0], (const v4f*)p[1],
        (const v4f*)p[2], (const v4f*)p[3], (const v4f*)p[4],
        (const v4f*)p[5], c[0], c[1], c[2], c[3], c[4], c[5], n4);
  };

  auto dyn = [&](const float* yin, float* kout) {
    gemm_qkv_k<<<gQKV, blk, 0, stream>>>(yin, Wq, Wk, Wv, bq, bk, bv,
                                         qb, kbuf, vbuf);
    attn_wmma_k<<<gA, blk, 0, stream>>>(qb, kbuf, vbuf, mask, attnO);
    gemm_wmma_k<<<gP, blk, 0, stream>>>(attnO, Wo, bo, att, 64, 64, 1, yin, zb);
    gemm_wmma_k<<<gF1, blk, 0, stream>>>(zb, W1, b1, h1, 64, 256, 2, nullptr, nullptr);
    gemm_wmma_k<<<gP, blk, 0, stream>>>(h1, W2, b2, kout, 256, 64, 3, att, nullptr);
  };

  // RKF56 (Fehlberg) stage coefficients a[s][i] and solution weights b[i]
  const float DT = 0.25f;
  const float ST[6][6] = {
      {0, 0, 0, 0, 0, 0},
      {0.25f, 0, 0, 0, 0, 0},
      {3.0f / 32.0f, 9.0f / 32.0f, 0, 0, 0, 0},
      {1932.0f / 2197.0f, -7200.0f / 2197.0f, 7296.0f / 2197.0f, 0, 0, 0},
      {439.0f / 216.0f, -8.0f, 3680.0f / 513.0f, -845.0f / 4104.0f, 0, 0},
      {-8.0f / 27.0f, 2.0f, -3544.0f / 2565.0f, 1859.0f / 4104.0f,
       -11.0f / 40.0f, 0}};
  const float BW[6] = {16.0f / 135.0f, 0.0f, 6656.0f / 12825.0f,
                       28561.0f / 56430.0f, -9.0f / 50.0f, 2.0f / 55.0f};

  // y = x  (all-zero coefficients -> pure copy)
  {
    float* pz[6] = {y, y, y, y, y, y};
    const float cz[6] = {0, 0, 0, 0, 0, 0};
    combo(y, x, pz, cz);
  }

  for (int step = 0; step < 4; ++step) {
    for (int s = 0; s < 6; ++s) {
      float c[6];
      for (int i = 0; i < 6; ++i) c[i] = DT * ST[s][i];
      combo(ystage, y, kst, c);
      dyn(ystage, kst[s]);
    }
    float c[6];
    for (int i = 0; i < 6; ++i) c[i] = DT * BW[i];
    float* outp = (step == 3) ? (float*)d_out : y;
    combo(outp, y, kst, c);
  }
}